// BranchRoute_15728170238619
// MI455X (gfx1250) — compile-verified
//
#include <hip/hip_runtime.h>

// ---------------------------------------------------------------------------
// BranchRoute fused kernel for MI455X (gfx1250, wave32).
//
// score = sigmoid(x @ W_gate + b_gate); hot = score > 0.5  <=>  z > 0 where
// z = x.W + b. Gate GEMM via V_WMMA_F32_16X16X4_F32 (exact fp32 path).
// Routing bits extracted with wave32 ballots; outputs streamed with
// nontemporal float4 stores. Bandwidth-bound: ~256 MB => ~11 us @ 23.3 TB/s.
//
// Trick: only D columns 0,1 are ever read (ballot bits 0,1,16,17), and WMMA
// dataflow is per-column, so B columns n>=2 may hold arbitrary finite data.
// Lanes with n>=2 load W column (n&1) -> no divergent branch, no EXEC
// save/restore in the 256-iteration hot loop.
// ---------------------------------------------------------------------------

typedef __attribute__((ext_vector_type(2))) float v2f;
typedef __attribute__((ext_vector_type(4))) float v4f;
typedef __attribute__((ext_vector_type(8))) float v8f;

#define D_MODEL     1024
#define N_BRANCHES  2
#define TILE_M      16
#define WAVES_PER_BLOCK 8

__global__ __launch_bounds__(32 * WAVES_PER_BLOCK)
void branch_route_kernel(const float* __restrict__ x,
                         const float* __restrict__ Wg,   // [D_MODEL, 2]
                         const float* __restrict__ bg,   // [2]
                         float* __restrict__ out0,
                         float* __restrict__ out1,
                         float* __restrict__ outc,
                         int n_tiles)
{
    const int lane = threadIdx.x & 31;
    const int wave = threadIdx.x >> 5;
    const int tile = blockIdx.x * WAVES_PER_BLOCK + wave;
    if (tile >= n_tiles) return;               // wave-uniform; EXEC stays all-1s

    const int row0  = tile * TILE_M;
    const int m     = lane & 15;               // A-matrix row / B-matrix col (N)
    const int khalf = (lane >> 4) * 2;         // K sub-offset: 0 (lanes 0-15) or 2

    // ---------------- Phase 1: gate GEMM, D = A(16x4) x B(4x16) + C ----------
    // A: x[row0+m][k+khalf+{0,1}] in {a.x, a.y}
    // B: W column (m&1), rows k+khalf+{0,1}. Columns n>=2 are dead in D, so
    //    duplicated column data there is harmless -> branch-free loads.
    v8f acc = {};
    const float* __restrict__ arow  = x  + (size_t)(row0 + m) * D_MODEL + khalf;
    const float* __restrict__ wbase = Wg + (size_t)khalf * N_BRANCHES + (m & 1);

    #pragma unroll 4
    for (int k = 0; k < D_MODEL; k += 4) {
        v2f a;
        a.x = arow[k + 0];
        a.y = arow[k + 1];

        v2f b;
        b.x = wbase[(size_t)k * N_BRANCHES];                // W[k+khalf  ][m&1]
        b.y = wbase[(size_t)k * N_BRANCHES + N_BRANCHES];   // W[k+khalf+1][m&1]

        // (neg_a, A, neg_b, B, c_mod, C, reuse_a, reuse_b)
        acc = __builtin_amdgcn_wmma_f32_16x16x4_f32(
            false, a, false, b, (short)0, acc, false, false);
    }

    // D layout (16x16 f32): vgpr j, lanes 0-15 -> (M=j,   N=lane),
    //                                lanes 16-31 -> (M=j+8, N=lane-16).
    // hot iff z + bias > 0 (sigmoid(z+b) > 0.5). Only lanes 0,1,16,17 matter.
    const float bias = (m < N_BRANCHES) ? bg[m] : 0.0f;

    unsigned hot0 = 0, hot1 = 0;               // bit r = token row0+r routed
    #pragma unroll
    for (int j = 0; j < 8; ++j) {
        const unsigned bal = __builtin_amdgcn_ballot_w32((acc[j] + bias) > 0.0f);
        hot0 |= ((bal >>  0) & 1u) << j;       // lane 0  : (M=j,   N=0)
        hot1 |= ((bal >>  1) & 1u) << j;       // lane 1  : (M=j,   N=1)
        hot0 |= ((bal >> 16) & 1u) << (j + 8); // lane 16 : (M=j+8, N=0)
        hot1 |= ((bal >> 17) & 1u) << (j + 8); // lane 17 : (M=j+8, N=1)
    }

    // ---------------- Phase 2: stream tile out with masks --------------------
    // Tile (64 KB) is L2-hot from phase 1; re-read coalesced as float4.
    // Outputs are write-once streams -> nontemporal stores (NT hint), keeping
    // L2 for the x re-read.
    #pragma unroll 2
    for (int r = 0; r < TILE_M; ++r) {
        const float m0 = ((hot0 >> r) & 1u) ? 1.0f : 0.0f;
        const float m1 = ((hot1 >> r) & 1u) ? 1.0f : 0.0f;
        const float mc = m0 + m1;              // 0, x, or 2x in combine slot

        const size_t base = (size_t)(row0 + r) * D_MODEL;
        const v4f* __restrict__ xr = reinterpret_cast<const v4f*>(x    + base);
        v4f* __restrict__ o0       = reinterpret_cast<v4f*>(out0 + base);
        v4f* __restrict__ o1       = reinterpret_cast<v4f*>(out1 + base);
        v4f* __restrict__ oc       = reinterpret_cast<v4f*>(outc + base);

        #pragma unroll
        for (int c = lane; c < D_MODEL / 4; c += 32) {
            const v4f v = xr[c];
            __builtin_nontemporal_store(v * m0, &o0[c]);
            __builtin_nontemporal_store(v * m1, &o1[c]);
            __builtin_nontemporal_store(v * mc, &oc[c]);
        }
    }
}

extern "C" void kernel_launch(void* const* d_in, const int* in_sizes, int n_in,
                              void* d_out, int out_size, void* d_ws, size_t ws_size,
                              hipStream_t stream) {
    const float* x  = (const float*)d_in[0];   // [N_TOKENS, D_MODEL]
    const float* Wg = (const float*)d_in[1];   // [D_MODEL, 2]
    const float* bg = (const float*)d_in[2];   // [2]

    const int n_tokens = in_sizes[0] / D_MODEL;
    const int n_tiles  = n_tokens / TILE_M;    // 1024 for N_TOKENS=16384

    const size_t plane = (size_t)n_tokens * D_MODEL;
    float* out0 = (float*)d_out;               // x_0
    float* out1 = out0 + plane;                // x_1
    float* outc = out1 + plane;                // x_comb

    const int blocks = (n_tiles + WAVES_PER_BLOCK - 1) / WAVES_PER_BLOCK;
    branch_route_kernel<<<blocks, 32 * WAVES_PER_BLOCK, 0, stream>>>(
        x, Wg, bg, out0, out1, outc, n_tiles);
}